// surfacePropLoss_34643206209565
// MI455X (gfx1250) — compile-verified
//
#include <hip/hip_runtime.h>
#include <hip/hip_bf16.h>

// ---------------------------------------------------------------------------
// Surface-property loss for point clouds, MI455X (gfx1250, wave32).
//
// d2(q,c) = |q|^2 + |c|^2 - 2 q.c ; the q.c term for each 16(query) x
// 16(candidate) tile is one V_WMMA_F32_16X16X4_F32 (K=3 padded to 4).
// One wave owns 16 queries (B fragment built once), streams candidates
// through LDS in chunks of 128 (8 WMMA sub-tiles per barrier, coalesced
// b128 loads + global_prefetch of the next chunk), keeps a per-lane
// register top-16 (each query split across lanes l / l^16 per the C/D
// layout), then merges via shuffles + rank-select.
// ---------------------------------------------------------------------------

typedef float v2f __attribute__((ext_vector_type(2)));
typedef float v8f __attribute__((ext_vector_type(8)));

#define KNN 16
#define CHUNK 128           // candidates staged per LDS round (8 sub-tiles)
#define W_NORMAL 1.0f
#define W_SURFVAR 1.0f

#if __has_builtin(__builtin_amdgcn_wmma_f32_16x16x4_f32)
#define HAVE_WMMA_F32X4 1
#else
#pragma message("wmma_f32_16x16x4_f32 builtin not available: scalar fallback in use")
#endif

// ---------------------------------------------------------------------------
// kNN kernel: one wave (32 threads) per 16 queries of one segment.
// segLen = 8192 (global) or 256 (patch); both are multiples of CHUNK.
// Neighbor ids written as u16 global point ids (max 32767 fits).
// ---------------------------------------------------------------------------
__global__ __launch_bounds__(32) void knn16_kernel(const float* __restrict__ pts,
                                                   unsigned short* __restrict__ idxOut,
                                                   int segLen)
{
    const int lane = threadIdx.x;            // 0..31
    const int tilesPerSeg = segLen >> 4;
    const int seg     = blockIdx.x / tilesPerSeg;
    const int qtile   = blockIdx.x % tilesPerSeg;
    const int segBase = seg * segLen;

    const int qcol = lane & 15;              // query column N in the WMMA tile
    const int hi   = lane >> 4;              // 0: rows M=0..7 / K={0,1}; 1: M=8..15 / K={2,3}

    // Load this lane's query point (column N of B-fragment), compute |q|^2.
    const int   qg = segBase + qtile * 16 + qcol;
    const float qx = pts[qg * 3 + 0];
    const float qy = pts[qg * 3 + 1];
    const float qz = pts[qg * 3 + 2];
    const float qn = qx * qx + qy * qy + qz * qz;

    // B fragment (4x16 f32): lanes 0-15 hold rows K={0,1}, lanes 16-31 rows K={2,3}.
    v2f bfrag;
    bfrag.x = hi ? qz : qx;
    bfrag.y = hi ? 0.0f : qy;

    // Per-lane top-16 (this lane sees candidates M = v + 8*hi of every tile).
    float bd[KNN];
    int   bi[KNN];
#pragma unroll
    for (int i = 0; i < KNN; ++i) { bd[i] = 3.4e38f; bi[i] = 0; }
    float worst = 3.4e38f;
    int   worstSlot = 0;

    __shared__ float4 sCand[CHUNK];          // xyz + |c|^2 of staged candidates

    for (int chunk = 0; chunk < segLen; chunk += CHUNK) {
        const int cbase = segBase + chunk;

        // Cooperative stage: 4 candidates per lane via 3 coalesced b128 loads
        // (48 contiguous bytes; cbase is a multiple of CHUNK so 16B-aligned).
        {
            const float4* src =
                reinterpret_cast<const float4*>(pts + (size_t)(cbase + lane * 4) * 3);
            const float4 A = src[0];
            const float4 Bv = src[1];
            const float4 C = src[2];
            sCand[lane * 4 + 0] =
                make_float4(A.x, A.y, A.z, A.x * A.x + A.y * A.y + A.z * A.z);
            sCand[lane * 4 + 1] =
                make_float4(A.w, Bv.x, Bv.y, A.w * A.w + Bv.x * Bv.x + Bv.y * Bv.y);
            sCand[lane * 4 + 2] =
                make_float4(Bv.z, Bv.w, C.x, Bv.z * Bv.z + Bv.w * Bv.w + C.x * C.x);
            sCand[lane * 4 + 3] =
                make_float4(C.y, C.z, C.w, C.y * C.y + C.z * C.z + C.w * C.w);
        }
        // Prefetch next chunk region into the WGP cache (global_prefetch_b8).
        if (chunk + CHUNK < segLen)
            __builtin_prefetch(pts + (size_t)(cbase + CHUNK + lane * 4) * 3, 0, 3);
        __syncthreads();

#pragma unroll 1
        for (int st = 0; st < CHUNK / 16; ++st) {
            // A fragment (16x4 f32): candidate row M = lane&15, K pair by lane half.
            const float4 cme = sCand[st * 16 + (lane & 15)];
            v2f afrag;
            afrag.x = hi ? cme.z : cme.x;
            afrag.y = hi ? 0.0f : cme.y;

            v8f dot = {0.f, 0.f, 0.f, 0.f, 0.f, 0.f, 0.f, 0.f};
#if defined(HAVE_WMMA_F32X4)
            // D[M][N] = sum_k A[M,k] * B[k,N]  (K=4, 4th dim zero-padded)
            dot = __builtin_amdgcn_wmma_f32_16x16x4_f32(
                /*neg_a=*/false, afrag, /*neg_b=*/false, bfrag,
                /*c_mod=*/(short)0, dot, /*reuse_a=*/false, /*reuse_b=*/false);
#else
#pragma unroll
            for (int v = 0; v < 8; ++v) {
                const float4 c = sCand[st * 16 + v + 8 * hi];
                dot[v] = c.x * qx + c.y * qy + c.z * qz;
            }
#endif
            // Lane l holds dots for query qcol with candidates M = v + 8*hi.
#pragma unroll
            for (int v = 0; v < 8; ++v) {
                const int   m   = v + 8 * hi;
                const float d2  = qn + sCand[st * 16 + m].w - 2.0f * dot[v];
                const int   gid = cbase + st * 16 + m;
                if (d2 < worst) {
                    // Replace current worst slot (static-indexed, stays in VGPRs).
#pragma unroll
                    for (int i = 0; i < KNN; ++i)
                        if (i == worstSlot) { bd[i] = d2; bi[i] = gid; }
                    // Recompute worst.
                    worst = bd[0]; worstSlot = 0;
#pragma unroll
                    for (int i = 1; i < KNN; ++i)
                        if (bd[i] > worst) { worst = bd[i]; worstSlot = i; }
                }
            }
        }
        __syncthreads();
    }

    // Merge lane l with partner lane l^16 (same query, other candidate half).
    float od[KNN];
    int   oi[KNN];
#pragma unroll
    for (int i = 0; i < KNN; ++i) {
        od[i] = __shfl(bd[i], lane ^ 16, 32);
        oi[i] = __shfl(bi[i], lane ^ 16, 32);
    }

    // Rank-select: element survives if fewer than 16 of the 32 are strictly
    // smaller under (dist, idx) order (matches top_k lowest-index tie-break).
    // Each lane writes only its own qualifying elements -> exactly 16 stores.
    unsigned short* qout = idxOut + (size_t)qg * KNN;
#pragma unroll
    for (int i = 0; i < KNN; ++i) {
        int rank = 0;
#pragma unroll
        for (int j = 0; j < KNN; ++j) {
            if (j != i)
                rank += ((bd[j] < bd[i]) || (bd[j] == bd[i] && bi[j] < bi[i])) ? 1 : 0;
            rank += ((od[j] < bd[i]) || (od[j] == bd[i] && oi[j] < bi[i])) ? 1 : 0;
        }
        if (rank < KNN) qout[rank] = (unsigned short)bi[i];
    }
}

// ---------------------------------------------------------------------------
// Per-point covariance + closed-form symmetric 3x3 eigensolver.
// Returns eigenvector of lambda_min (sign-ambiguous) and sv = lmin/trace.
// ---------------------------------------------------------------------------
__device__ __forceinline__ void covEig(const float* __restrict__ pts,
                                       const unsigned short* __restrict__ nidx,
                                       float px, float py, float pz,
                                       float nrm[3], float* sv)
{
    float c00 = 0.f, c01 = 0.f, c02 = 0.f, c11 = 0.f, c12 = 0.f, c22 = 0.f;
#pragma unroll
    for (int k = 0; k < KNN; ++k) {
        const int id = nidx[k];
        const float x = pts[id * 3 + 0] - px;
        const float y = pts[id * 3 + 1] - py;
        const float z = pts[id * 3 + 2] - pz;
        c00 += x * x; c01 += x * y; c02 += x * z;
        c11 += y * y; c12 += y * z; c22 += z * z;
    }
    const float tr = c00 + c11 + c22;
    const float q  = tr * (1.0f / 3.0f);
    const float p1 = c01 * c01 + c02 * c02 + c12 * c12;
    const float a00 = c00 - q, a11 = c11 - q, a22 = c22 - q;
    const float p2 = a00 * a00 + a11 * a11 + a22 * a22 + 2.0f * p1;
    const float p  = sqrtf(fmaxf(p2, 1e-30f) * (1.0f / 6.0f));
    const float ip = 1.0f / p;
    const float b00 = a00 * ip, b01 = c01 * ip, b02 = c02 * ip;
    const float b11 = a11 * ip, b12 = c12 * ip, b22 = a22 * ip;
    const float detB = b00 * (b11 * b22 - b12 * b12)
                     - b01 * (b01 * b22 - b12 * b02)
                     + b02 * (b01 * b12 - b11 * b02);
    const float r   = fminf(fmaxf(0.5f * detB, -1.0f), 1.0f);
    const float phi = acosf(r) * (1.0f / 3.0f);
    const float lmin = q + 2.0f * p * cosf(phi + 2.0943951023931953f); // +2pi/3

    // Eigenvector of lmin: largest cross product of rows of (C - lmin*I).
    const float r0x = c00 - lmin, r0y = c01,        r0z = c02;
    const float r1x = c01,        r1y = c11 - lmin, r1z = c12;
    const float r2x = c02,        r2y = c12,        r2z = c22 - lmin;
    float v0x = r0y * r1z - r0z * r1y, v0y = r0z * r1x - r0x * r1z, v0z = r0x * r1y - r0y * r1x;
    float v1x = r0y * r2z - r0z * r2y, v1y = r0z * r2x - r0x * r2z, v1z = r0x * r2y - r0y * r2x;
    float v2x = r1y * r2z - r1z * r2y, v2y = r1z * r2x - r1x * r2z, v2z = r1x * r2y - r1y * r2x;
    const float n0 = v0x * v0x + v0y * v0y + v0z * v0z;
    const float n1 = v1x * v1x + v1y * v1y + v1z * v1z;
    const float n2 = v2x * v2x + v2y * v2y + v2z * v2z;
    float vx = v0x, vy = v0y, vz = v0z, nn = n0;
    if (n1 > nn) { vx = v1x; vy = v1y; vz = v1z; nn = n1; }
    if (n2 > nn) { vx = v2x; vy = v2y; vz = v2z; nn = n2; }
    const float inv = rsqrtf(fmaxf(nn, 1e-30f));
    nrm[0] = vx * inv; nrm[1] = vy * inv; nrm[2] = vz * inv;
    *sv = lmin / fmaxf(tr, 1e-30f);
}

// ---------------------------------------------------------------------------
// Loss kernel: one thread per point; wave-reduce then atomicAdd.
// ---------------------------------------------------------------------------
__global__ __launch_bounds__(256) void props_kernel(const float* __restrict__ pts,
                                                    const unsigned short* __restrict__ idxG,
                                                    const unsigned short* __restrict__ idxP,
                                                    float* __restrict__ accum,
                                                    int total)
{
    const int g = blockIdx.x * blockDim.x + threadIdx.x;
    float ln = 0.0f, lsv = 0.0f;
    if (g < total) {
        const float px = pts[g * 3 + 0];
        const float py = pts[g * 3 + 1];
        const float pz = pts[g * 3 + 2];
        float ng[3], np3[3], svg, svp;
        covEig(pts, idxG + (size_t)g * KNN, px, py, pz, ng, &svg);
        covEig(pts, idxP + (size_t)g * KNN, px, py, pz, np3, &svp);
        const float dx = fabsf(np3[0]) - fabsf(ng[0]);
        const float dy = fabsf(np3[1]) - fabsf(ng[1]);
        const float dz = fabsf(np3[2]) - fabsf(ng[2]);
        ln = sqrtf(dx * dx + dy * dy + dz * dz);
        const float dsv = svp - svg;
        lsv = dsv * dsv;
    }
    // wave32 tree reduction
#pragma unroll
    for (int off = 16; off > 0; off >>= 1) {
        ln  += __shfl_down(ln, off, 32);
        lsv += __shfl_down(lsv, off, 32);
    }
    if ((threadIdx.x & 31) == 0) {
        atomicAdd(&accum[0], ln);
        atomicAdd(&accum[1], lsv);
    }
}

__global__ void finalize_kernel(const float* __restrict__ accum,
                                float* __restrict__ out, float invTotal)
{
    if (threadIdx.x == 0) {
        out[0] = accum[0] * invTotal * W_NORMAL;
        out[1] = accum[1] * invTotal * W_SURFVAR;
    }
}

// ---------------------------------------------------------------------------
extern "C" void kernel_launch(void* const* d_in, const int* in_sizes, int n_in,
                              void* d_out, int out_size, void* d_ws, size_t ws_size,
                              hipStream_t stream)
{
    const float* pts = (const float*)d_in[0];
    const int total = in_sizes[0] / 3;       // B*N = 32768
    const int B = 4;
    const int N = total / B;                 // 8192
    const int PATCH = N / 32;                // 256 points per patch

    unsigned short* idxG  = (unsigned short*)d_ws;
    unsigned short* idxP  = idxG + (size_t)total * KNN;
    float*          accum = (float*)(idxP + (size_t)total * KNN);

    hipMemsetAsync(accum, 0, 2 * sizeof(float), stream);

    // Global kNN: segment = one batch of N points.
    knn16_kernel<<<B * (N / 16), 32, 0, stream>>>(pts, idxG, N);
    // Patch kNN: segment = one patch of 256 points.
    knn16_kernel<<<total / 16, 32, 0, stream>>>(pts, idxP, PATCH);
    // Covariance + eigensolve + loss reduction.
    props_kernel<<<(total + 255) / 256, 256, 0, stream>>>(pts, idxG, idxP, accum, total);
    finalize_kernel<<<1, 32, 0, stream>>>(accum, (float*)d_out, 1.0f / (float)total);
}